// GraphEncoder_39247411151302
// MI455X (gfx1250) — compile-verified
//
#include <hip/hip_runtime.h>
#include <hip/hip_bf16.h>

typedef float v2f __attribute__((ext_vector_type(2)));
typedef float v8f __attribute__((ext_vector_type(8)));

#define FEAT 64  // hidden/output feature width

// ---------------- degree kernels ----------------
__global__ void deg_init(float* __restrict__ deg, int n) {
    int i = blockIdx.x * blockDim.x + threadIdx.x;
    if (i < n) deg[i] = 1.0f;  // self-loop
}

__global__ void deg_count(const long long* __restrict__ dst, float* __restrict__ deg, int e) {
    int i = blockIdx.x * blockDim.x + threadIdx.x;
    if (i < e) atomicAdd(&deg[dst[i]], 1.0f);
}

__global__ void deg_rsqrt(float* __restrict__ deg, int n) {
    int i = blockIdx.x * blockDim.x + threadIdx.x;
    if (i < n) deg[i] = rsqrtf(deg[i]);  // deg >= 1 always (self-loops)
}

// ---------------- WMMA GEMM: hh = (X @ W) * dinv[row]; write to H and ACC ----------------
// X: [N, K] row-major, W: [K, 64] row-major, H/ACC: [N, 64].
// Block = 128 threads = 4 waves; each wave computes one 16x16 tile of C;
// wave w covers output cols [16w, 16w+16). Grid.x = ceil(N/16) row tiles.
__global__ void gemm_wmma_f32(const float* __restrict__ X, const float* __restrict__ W,
                              const float* __restrict__ dinv,
                              float* __restrict__ H, float* __restrict__ ACC,
                              int N, int K) {
    const int wave = threadIdx.x >> 5;
    const int lane = threadIdx.x & 31;
    const int lmod = lane & 15;
    const int lhi  = lane >> 4;          // 0 for lanes 0-15, 1 for lanes 16-31
    const int row0 = blockIdx.x * 16;
    const int col0 = wave * 16;

    // clamp load row for (theoretical) ragged tail; N=100000 is a multiple of 16
    const int arow = min(row0 + lmod, N - 1);
    const float* __restrict__ xrow = X + (size_t)arow * K;

    v8f c = {0.f, 0.f, 0.f, 0.f, 0.f, 0.f, 0.f, 0.f};

    for (int k0 = 0; k0 < K; k0 += 4) {
        // A fragment: lane holds A[M=lmod][K = k0 + 2*lhi + {0,1}]  (contiguous pair)
        const int ka = k0 + 2 * lhi;
        v2f a;
        a.x = xrow[ka];
        a.y = xrow[ka + 1];
        // B fragment: lane holds B[K = k0 + 2*lhi + {0,1}][N = col0 + lmod]
        const float* __restrict__ bp = W + (size_t)ka * FEAT + col0 + lmod;
        v2f b;
        b.x = bp[0];
        b.y = bp[FEAT];
        // V_WMMA_F32_16X16X4_F32: full f32 matrix op, accumulate in c
        c = __builtin_amdgcn_wmma_f32_16x16x4_f32(
                /*neg_a=*/false, a, /*neg_b=*/false, b,
                /*c_mod=*/(short)0, c, /*reuse_a=*/false, /*reuse_b=*/false);
    }

    // Epilogue: C layout -> row = row0 + r + 8*lhi, col = col0 + lmod.
    // Scale by dinv[row]; write to H (gather source) and ACC (self-loop init).
#pragma unroll
    for (int r = 0; r < 8; ++r) {
        const int row = row0 + r + 8 * lhi;
        if (row < N) {
            const float v = c[r] * dinv[row];
            const size_t idx = (size_t)row * FEAT + col0 + lmod;
            H[idx]   = v;
            ACC[idx] = v;
        }
    }
}

// ---------------- edge scatter: ACC[dst] += H[src] ----------------
// 16 lanes per edge, each lane handles 4 consecutive floats (float4 read, 4 f32 atomics).
__global__ void edge_scatter(const long long* __restrict__ src, const long long* __restrict__ dst,
                             const float* __restrict__ H, float* __restrict__ ACC, int E) {
    const int t = blockIdx.x * blockDim.x + threadIdx.x;
    const int e = t >> 4;
    if (e >= E) return;
    const int p = (t & 15) << 2;
    const long long s = src[e];
    const long long d = dst[e];
    const float4 v = *reinterpret_cast<const float4*>(H + (size_t)s * FEAT + p);
    float* __restrict__ ap = ACC + (size_t)d * FEAT + p;
    atomicAdd(ap + 0, v.x);
    atomicAdd(ap + 1, v.y);
    atomicAdd(ap + 2, v.z);
    atomicAdd(ap + 3, v.w);
}

// ---------------- finalize: out = f(ACC*dinv + b) per row ----------------
// One wave (32 lanes) per 64-float row; 8 rows per 256-thread block.
// DO_RELU/DO_LN: relu then layernorm (layers 1,2); plain for layer 3.
template <bool DO_RELU_LN>
__global__ void row_finalize(const float* __restrict__ ACC, const float* __restrict__ dinv,
                             const float* __restrict__ bias, const float* __restrict__ gamma,
                             const float* __restrict__ beta, float* __restrict__ OUT, int N) {
    const int lane = threadIdx.x & 31;
    const int row  = blockIdx.x * 8 + (threadIdx.x >> 5);
    if (row >= N) return;
    const float* __restrict__ ar = ACC + (size_t)row * FEAT;
    const float di = dinv[row];
    float a0 = ar[lane]      * di + bias[lane];
    float a1 = ar[lane + 32] * di + bias[lane + 32];
    if (DO_RELU_LN) {
        a0 = fmaxf(a0, 0.f);
        a1 = fmaxf(a1, 0.f);
        float s = a0 + a1;
        float q = a0 * a0 + a1 * a1;
#pragma unroll
        for (int off = 16; off >= 1; off >>= 1) {
            s += __shfl_xor(s, off, 32);
            q += __shfl_xor(q, off, 32);
        }
        const float mean = s * (1.0f / FEAT);
        const float var  = q * (1.0f / FEAT) - mean * mean;
        const float rstd = rsqrtf(var + 1e-5f);
        a0 = (a0 - mean) * rstd * gamma[lane]      + beta[lane];
        a1 = (a1 - mean) * rstd * gamma[lane + 32] + beta[lane + 32];
    }
    float* __restrict__ orow = OUT + (size_t)row * FEAT;
    orow[lane]      = a0;
    orow[lane + 32] = a1;
}

// ---------------- orchestration ----------------
extern "C" void kernel_launch(void* const* d_in, const int* in_sizes, int n_in,
                              void* d_out, int out_size, void* d_ws, size_t ws_size,
                              hipStream_t stream) {
    const int NUM_IN = 128;
    const int N = in_sizes[0] / NUM_IN;      // 100000
    const int E = in_sizes[1] / 2;           // 1600000

    const float*     x     = (const float*)d_in[0];
    const long long* edge  = (const long long*)d_in[1];   // int64 [2, E]
    const long long* src   = edge;
    const long long* dstp  = edge + E;
    const float* W_in   = (const float*)d_in[2];
    const float* b_in   = (const float*)d_in[3];
    const float* g_in   = (const float*)d_in[4];
    const float* be_in  = (const float*)d_in[5];
    const float* W_h    = (const float*)d_in[6];
    const float* b_h    = (const float*)d_in[7];
    const float* g_h    = (const float*)d_in[8];
    const float* be_h   = (const float*)d_in[9];
    const float* W_out  = (const float*)d_in[10];
    const float* b_out  = (const float*)d_in[11];

    // workspace layout: dinv[N] | buf0[N*64] | buf1[N*64] (H) | buf2[N*64] (ACC)
    float* dinv = (float*)d_ws;
    float* buf0 = dinv + N;
    float* buf1 = buf0 + (size_t)N * FEAT;
    float* buf2 = buf1 + (size_t)N * FEAT;

    const int TB = 256;
    const int nBlkN   = (N + TB - 1) / TB;
    const int nBlkE   = (E + TB - 1) / TB;
    const int gemmBlk = (N + 15) / 16;
    const int scatBlk = ((E * 16) + TB - 1) / TB;
    const int rowBlk  = (N + 7) / 8;

    // degree -> dinv (shared across all three layers)
    deg_init <<<nBlkN, TB, 0, stream>>>(dinv, N);
    deg_count<<<nBlkE, TB, 0, stream>>>(dstp, dinv, E);
    deg_rsqrt<<<nBlkN, TB, 0, stream>>>(dinv, N);

    // ---- layer 1: x[ N,128 ] @ W_in -> relu -> LN ----
    gemm_wmma_f32<<<gemmBlk, 128, 0, stream>>>(x, W_in, dinv, buf1, buf2, N, NUM_IN);
    edge_scatter <<<scatBlk, TB, 0, stream>>>(src, dstp, buf1, buf2, E);
    row_finalize<true><<<rowBlk, TB, 0, stream>>>(buf2, dinv, b_in, g_in, be_in, buf0, N);

    // ---- layer 2: buf0[N,64] @ W_h -> relu -> LN ----
    gemm_wmma_f32<<<gemmBlk, 128, 0, stream>>>(buf0, W_h, dinv, buf1, buf2, N, FEAT);
    edge_scatter <<<scatBlk, TB, 0, stream>>>(src, dstp, buf1, buf2, E);
    row_finalize<true><<<rowBlk, TB, 0, stream>>>(buf2, dinv, b_h, g_h, be_h, buf0, N);

    // ---- layer 3: buf0[N,64] @ W_out -> d_out ----
    gemm_wmma_f32<<<gemmBlk, 128, 0, stream>>>(buf0, W_out, dinv, buf1, buf2, N, FEAT);
    edge_scatter <<<scatBlk, TB, 0, stream>>>(src, dstp, buf1, buf2, E);
    row_finalize<false><<<rowBlk, TB, 0, stream>>>(buf2, dinv, b_out, nullptr, nullptr,
                                                  (float*)d_out, N);
}